// U_S_encoder_12137577578912
// MI455X (gfx1250) — compile-verified
//
#include <hip/hip_runtime.h>
#include <hip/hip_bf16.h>

typedef __attribute__((ext_vector_type(2))) float v2f;
typedef __attribute__((ext_vector_type(8))) float v8f;

#define N_NODES 100000
#define N_EDGES 3200000
#define FDIM    256
#define BN_EPS  1e-5f
#define LDS_STRIDE 260   // 256 + 4 pad: lane row-stride hits distinct banks

// ---------------- degree / normalization ----------------
__global__ void k_deg_init(float* deg) {
    int i = blockIdx.x * blockDim.x + threadIdx.x;
    if (i < N_NODES) deg[i] = 1.0f;             // self-loop contributes 1
}

__global__ void k_deg_accum(const int* __restrict__ dst, float* deg) {
    int e = blockIdx.x * blockDim.x + threadIdx.x;
    if (e < N_EDGES) unsafeAtomicAdd(&deg[dst[e]], 1.0f);
}

__global__ void k_dinv(float* deg) {
    int i = blockIdx.x * blockDim.x + threadIdx.x;
    if (i < N_NODES) deg[i] = rsqrtf(deg[i]);   // deg >= 1 always (self-loops)
}

// ---------------- dense GEMM: Y[N,256] = X[N,256] @ W[256,256] ----------------
// block = 128 threads = 4 waves; block tile = 16 rows x 64 cols; wave tile = 16x16.
__global__ __launch_bounds__(128) void k_gemm_wmma(const float* __restrict__ X,
                                                   const float* __restrict__ W,
                                                   float* __restrict__ Y) {
    __shared__ float As[16 * LDS_STRIDE];
    const int m0 = blockIdx.x * 16;

    // cooperative load of the 16x256 X tile (16 KB) into LDS, float4 per thread x8
#pragma unroll
    for (int i = 0; i < 8; ++i) {
        int f   = threadIdx.x + 128 * i;   // float4 index 0..1023
        int row = f >> 6;                  // 64 float4 per row
        int c4  = f & 63;
        float4 v = ((const float4*)(X + (size_t)(m0 + row) * FDIM))[c4];
        *(float4*)&As[row * LDS_STRIDE + c4 * 4] = v;
    }
    __syncthreads();

    const int wave = threadIdx.x >> 5;     // wave32
    const int lane = threadIdx.x & 31;
    const int lo   = lane & 15;
    const int hi   = lane >> 4;            // 0: K pair {k,k+1}, 1: {k+2,k+3}
    const int n0   = blockIdx.y * 64 + wave * 16;

    v8f acc = {};
    const float* wcol = W + n0 + lo;                       // column base in W
    const float* arow = &As[lo * LDS_STRIDE + 2 * hi];     // A row base in LDS

#pragma unroll 4
    for (int k = 0; k < FDIM; k += 4) {
        // A 16x4 fragment: lanes 0-15 hold M=lane, VGPR j = K=k+2*hi+j
        v2f a = *(const v2f*)(arow + k);
        // B 4x16 fragment: lanes 0-15 hold N=lo, VGPR j = K=k+2*hi+j
        int kk = k + 2 * hi;
        v2f b;
        b.x = wcol[(size_t)kk * FDIM];
        b.y = wcol[(size_t)(kk + 1) * FDIM];
        acc = __builtin_amdgcn_wmma_f32_16x16x4_f32(
            /*neg_a=*/false, a, /*neg_b=*/false, b,
            /*c_mod=*/(short)0, acc, /*reuse_a=*/false, /*reuse_b=*/false);
    }

    // C/D layout: VGPR r -> row M=r (lanes 0-15) / M=r+8 (lanes 16-31), N=lo
    float* yc = Y + (size_t)m0 * FDIM + n0 + lo;
#pragma unroll
    for (int r = 0; r < 8; ++r)
        yc[(size_t)(r + 8 * hi) * FDIM] = acc[r];
}

// ---------------- aggregation: self-loop init + bias ----------------
__global__ void k_agg_init(const float* __restrict__ Hd, const float* __restrict__ dinv,
                           const float* __restrict__ bias, float* __restrict__ out) {
    long long t = (long long)blockIdx.x * blockDim.x + threadIdx.x;  // float4 idx
    if (t >= (long long)N_NODES * (FDIM / 4)) return;
    int row = (int)(t >> 6);
    int c4  = (int)(t & 63);
    float dn = dinv[row];
    dn *= dn;                                  // self-loop norm = dinv[i]^2
    float4 h = ((const float4*)Hd)[t];
    float4 b = ((const float4*)bias)[c4];
    float4 o = { h.x * dn + b.x, h.y * dn + b.y, h.z * dn + b.z, h.w * dn + b.w };
    ((float4*)out)[t] = o;
}

// ---------------- aggregation: one wave32 per edge, scatter-add ----------------
__global__ __launch_bounds__(256) void k_agg_edges(const float* __restrict__ H,
                                                   const int* __restrict__ src,
                                                   const int* __restrict__ dst,
                                                   const float* __restrict__ dinv,
                                                   float* __restrict__ out) {
    long long w = ((long long)blockIdx.x * blockDim.x + threadIdx.x) >> 5;
    if (w >= N_EDGES) return;
    int lane = threadIdx.x & 31;
    int s = src[w];
    int d = dst[w];
    float nrm = dinv[s] * dinv[d];
    const float4* hp = (const float4*)(H + (size_t)s * FDIM);
    float* op = out + (size_t)d * FDIM;
#pragma unroll
    for (int j = 0; j < 2; ++j) {
        int c4 = lane + 32 * j;
        float4 v = hp[c4];                     // L2-resident gather (h fits in 192MB L2)
        unsafeAtomicAdd(op + c4 * 4 + 0, v.x * nrm);
        unsafeAtomicAdd(op + c4 * 4 + 1, v.y * nrm);
        unsafeAtomicAdd(op + c4 * 4 + 2, v.z * nrm);
        unsafeAtomicAdd(op + c4 * 4 + 3, v.w * nrm);
    }
}

// ---------------- batch norm (training-mode stats) + ReLU ----------------
__global__ void k_zero256(float* a, float* b) {
    int f = threadIdx.x;
    a[f] = 0.0f;
    b[f] = 0.0f;
}

__global__ __launch_bounds__(256) void k_bn_reduce(const float* __restrict__ h,
                                                   float* __restrict__ sum,
                                                   float* __restrict__ sumsq) {
    int f  = threadIdx.x;                 // feature column (coalesced across threads)
    int r0 = blockIdx.x * 256;
    int r1 = min(r0 + 256, N_NODES);
    float s = 0.0f, ss = 0.0f;
    for (int r = r0; r < r1; ++r) {
        float v = h[(size_t)r * FDIM + f];
        s  += v;
        ss += v * v;
    }
    unsafeAtomicAdd(&sum[f], s);
    unsafeAtomicAdd(&sumsq[f], ss);
}

__global__ void k_bn_finalize(const float* __restrict__ sum, const float* __restrict__ sumsq,
                              const float* __restrict__ gamma, const float* __restrict__ beta,
                              float* __restrict__ scale, float* __restrict__ shift) {
    int f = threadIdx.x;
    const float invN = 1.0f / (float)N_NODES;
    float mean = sum[f] * invN;
    float var  = sumsq[f] * invN - mean * mean;
    float istd = rsqrtf(var + BN_EPS);
    float sc   = istd * gamma[f];
    scale[f] = sc;
    shift[f] = beta[f] - mean * sc;
}

__global__ void k_bn_relu(float* __restrict__ h, const float* __restrict__ scale,
                          const float* __restrict__ shift) {
    long long t = (long long)blockIdx.x * blockDim.x + threadIdx.x;  // float4 idx
    if (t >= (long long)N_NODES * (FDIM / 4)) return;
    int c4 = (int)(t & 63);
    float4 v  = ((float4*)h)[t];
    float4 sc = ((const float4*)scale)[c4];
    float4 sh = ((const float4*)shift)[c4];
    v.x = fmaxf(v.x * sc.x + sh.x, 0.0f);
    v.y = fmaxf(v.y * sc.y + sh.y, 0.0f);
    v.z = fmaxf(v.z * sc.z + sh.z, 0.0f);
    v.w = fmaxf(v.w * sc.w + sh.w, 0.0f);
    ((float4*)h)[t] = v;
}

// ---------------- driver ----------------
extern "C" void kernel_launch(void* const* d_in, const int* in_sizes, int n_in,
                              void* d_out, int out_size, void* d_ws, size_t ws_size,
                              hipStream_t stream) {
    const float* x     = (const float*)d_in[0];
    const int*   ei    = (const int*)  d_in[1];   // [2, E]: row0=src, row1=dst
    const float* W1    = (const float*)d_in[2];
    const float* b1    = (const float*)d_in[3];
    const float* Wmu   = (const float*)d_in[4];
    const float* bmu   = (const float*)d_in[5];
    const float* Wls   = (const float*)d_in[6];
    const float* bls   = (const float*)d_in[7];
    const float* gamma = (const float*)d_in[8];
    const float* beta  = (const float*)d_in[9];

    const int* src = ei;
    const int* dst = ei + N_EDGES;

    float* mu = (float*)d_out;
    float* ls = (float*)d_out + (size_t)N_NODES * FDIM;

    char* ws = (char*)d_ws;
    const size_t featBytes = (size_t)N_NODES * FDIM * sizeof(float);  // 102.4 MB
    float* bufA  = (float*)ws; ws += featBytes;        // dense GEMM output
    float* bufB  = (float*)ws; ws += featBytes;        // aggregated hidden (BN in place)
    float* dinv  = (float*)ws; ws += ((size_t)N_NODES * sizeof(float) + 255) & ~(size_t)255;
    float* sum   = (float*)ws; ws += 1024;
    float* sumsq = (float*)ws; ws += 1024;
    float* scale = (float*)ws; ws += 1024;
    float* shift = (float*)ws; ws += 1024;

    const dim3 gemmGrid(N_NODES / 16, FDIM / 64);      // 6250 x 4
    const int  elem4   = N_NODES * (FDIM / 4);         // 6.4M float4
    const int  eBlocks = N_EDGES / 8;                  // 1 wave32 per edge, 8 waves/block

    // degrees -> dinv (in place)
    k_deg_init <<<(N_NODES + 255) / 256, 256, 0, stream>>>(dinv);
    k_deg_accum<<<(N_EDGES + 255) / 256, 256, 0, stream>>>(dst, dinv);
    k_dinv     <<<(N_NODES + 255) / 256, 256, 0, stream>>>(dinv);

    // layer 1: h = A_hat @ (x @ W1) + b1
    k_gemm_wmma<<<gemmGrid, 128, 0, stream>>>(x, W1, bufA);
    k_agg_init <<<(elem4 + 255) / 256, 256, 0, stream>>>(bufA, dinv, b1, bufB);
    k_agg_edges<<<eBlocks, 256, 0, stream>>>(bufA, src, dst, dinv, bufB);

    // BN (training stats) + ReLU, in place on bufB
    k_zero256    <<<1, 256, 0, stream>>>(sum, sumsq);
    k_bn_reduce  <<<(N_NODES + 255) / 256, 256, 0, stream>>>(bufB, sum, sumsq);
    k_bn_finalize<<<1, 256, 0, stream>>>(sum, sumsq, gamma, beta, scale, shift);
    k_bn_relu    <<<(elem4 + 255) / 256, 256, 0, stream>>>(bufB, scale, shift);

    // mu head
    k_gemm_wmma<<<gemmGrid, 128, 0, stream>>>(bufB, Wmu, bufA);
    k_agg_init <<<(elem4 + 255) / 256, 256, 0, stream>>>(bufA, dinv, bmu, mu);
    k_agg_edges<<<eBlocks, 256, 0, stream>>>(bufA, src, dst, dinv, mu);

    // logstd head
    k_gemm_wmma<<<gemmGrid, 128, 0, stream>>>(bufB, Wls, bufA);
    k_agg_init <<<(elem4 + 255) / 256, 256, 0, stream>>>(bufA, dinv, bls, ls);
    k_agg_edges<<<eBlocks, 256, 0, stream>>>(bufA, src, dst, dinv, ls);
}